// MSGMS_Loss_15753940042119
// MI455X (gfx1250) — compile-verified
//
#include <hip/hip_runtime.h>
#include <cstdint>
#include <cstddef>

// ---------------------------------------------------------------------------
// MS-GMS loss for MI455X (gfx1250, wave32).
// Memory-bound pipeline: gray -> pooled gray pyramid -> per-scale tiled
// median3x3 + Prewitt + GMS with async direct-to-LDS halo loads, WMMA-based
// deterministic reductions.
// ---------------------------------------------------------------------------

#ifndef __has_builtin
#define __has_builtin(x) 0
#endif

#if __has_builtin(__builtin_amdgcn_global_load_async_to_lds_b32)
#define HAVE_ASYNC_LDS 1
#else
#define HAVE_ASYNC_LDS 0
#endif

#if __has_builtin(__builtin_amdgcn_wmma_f32_16x16x4_f32)
#define HAVE_WMMA_F32 1
#else
#define HAVE_WMMA_F32 0
#endif

#if defined(__HIP_DEVICE_COMPILE__) && !HAVE_ASYNC_LDS
#warning "gfx1250 device pass: __builtin_amdgcn_global_load_async_to_lds_b32 NOT available - using VGPR-staged fallback"
#endif
#if defined(__HIP_DEVICE_COMPILE__) && !HAVE_WMMA_F32
#warning "gfx1250 device pass: __builtin_amdgcn_wmma_f32_16x16x4_f32 NOT available - using shuffle fallback"
#endif

// Builtin wants address-space-qualified int* operands (clang diagnostic:
// parameter type '__device__ int *').
typedef __attribute__((address_space(1))) int glb_i32_t;
typedef __attribute__((address_space(3))) int lds_i32_t;

typedef __attribute__((ext_vector_type(2))) float v2f;
typedef __attribute__((ext_vector_type(8))) float v8f;

#define C_GMS 0.0026f

// ---------------- async global->LDS helpers ----------------

__device__ __forceinline__ void async_ld_f32(float* lds_dst, const float* gsrc) {
#if defined(__HIP_DEVICE_COMPILE__) && HAVE_ASYNC_LDS
  __builtin_amdgcn_global_load_async_to_lds_b32(
      (glb_i32_t*)gsrc, (lds_i32_t*)lds_dst, /*offset=*/0, /*cpol=*/0);
#else
  *lds_dst = *gsrc;
#endif
}

__device__ __forceinline__ void wait_async_lds() {
#if defined(__HIP_DEVICE_COMPILE__)
#if __has_builtin(__builtin_amdgcn_s_wait_asynccnt)
  __builtin_amdgcn_s_wait_asynccnt(0);
#endif
#endif
}

// ---------------- WMMA-assisted wave32 / block reductions ----------------
// V_WMMA_F32_16X16X4_F32 row-sum trick with B == ones(4x16):
//   A layout (16x4 f32, 2 VGPRs): lane L<16 holds A[M=L][K=0..1],
//   lane L>=16 holds A[M=L-16][K=2..3]. Set A = (v, 0) per lane so
//   rowsum(M) = v[lane M] + v[lane M+16]. D[M][N] = rowsum(M) for all N.
//   D layout: VGPR r = rows M=r (lanes 0-15) / M=8+r (lanes 16-31), so
//   sum of the 8 D VGPRs in a lane = sum over half the rows; one xor-16
//   shuffle completes the full 32-lane sum (identical in every lane).
__device__ __forceinline__ float wave_sum32(float v) {
#if defined(__HIP_DEVICE_COMPILE__) && HAVE_WMMA_F32
  v2f a; a[0] = v;    a[1] = 0.0f;
  v2f b; b[0] = 1.0f; b[1] = 1.0f;
  v8f c = {0.f, 0.f, 0.f, 0.f, 0.f, 0.f, 0.f, 0.f};
  c = __builtin_amdgcn_wmma_f32_16x16x4_f32(false, a, false, b,
                                            (short)0, c, false, false);
  float t = ((c[0] + c[1]) + (c[2] + c[3])) + ((c[4] + c[5]) + (c[6] + c[7]));
  return t + __shfl_xor(t, 16, 32);
#else
  for (int off = 16; off > 0; off >>= 1) v += __shfl_xor(v, off, 32);
  return v;
#endif
}

// Valid result only on threadIdx.x == 0. Block must be 256 threads.
__device__ __forceinline__ float block_sum256(float v, float* sRed8) {
  float w = wave_sum32(v);                    // uniform across wave (EXEC all-1s)
  const int lane = threadIdx.x & 31;
  const int wid  = threadIdx.x >> 5;
  if (lane == 0) sRed8[wid] = w;
  __syncthreads();
  float total = 0.f;
  if (threadIdx.x == 0) {
#pragma unroll
    for (int i = 0; i < 8; ++i) total += sRed8[i];
  }
  return total;
}

// ---------------- small math helpers ----------------

__device__ __forceinline__ void mm_(float& a, float& b) {
  float lo = fminf(a, b);
  b = fmaxf(a, b);
  a = lo;
}

__device__ __forceinline__ float median9(float p0, float p1, float p2,
                                         float p3, float p4, float p5,
                                         float p6, float p7, float p8) {
  // 19 compare-exchange median-of-9 network (Paeth).
  mm_(p1, p2); mm_(p4, p5); mm_(p7, p8);
  mm_(p0, p1); mm_(p3, p4); mm_(p6, p7);
  mm_(p1, p2); mm_(p4, p5); mm_(p7, p8);
  mm_(p0, p3); mm_(p5, p8); mm_(p4, p7);
  mm_(p3, p6); mm_(p1, p4); mm_(p2, p5);
  mm_(p4, p7); mm_(p4, p2); mm_(p6, p4);
  mm_(p4, p2);
  return p4;
}

// m points at median-plane element (oi,oj); stride 35.
__device__ __forceinline__ float prewitt_mag(const float* m) {
  float t00 = m[0],  t01 = m[1],  t02 = m[2];
  float t10 = m[35],               t12 = m[37];
  float t20 = m[70], t21 = m[71], t22 = m[72];
  float gx = ((t02 + t12 + t22) - (t00 + t10 + t20)) * (1.f / 3.f);
  float gy = ((t00 + t01 + t02) - (t20 + t21 + t22)) * (1.f / 3.f);
  return sqrtf(gx * gx + gy * gy);
}

// ---------------- kernel 1: RGB -> gray (channel mean), both images ----------------
// grid.x * 256 threads * 4 pixels = 16*512*512 pixels; grid.y selects image.
__global__ __launch_bounds__(256) void gray_kernel(
    const float* __restrict__ inI, const float* __restrict__ inR,
    float* __restrict__ gI, float* __restrict__ gR) {
  const float* src = blockIdx.y ? inR : inI;
  float*       dst = blockIdx.y ? gR : gI;
  const size_t HW = 512u * 512u;
  size_t p = ((size_t)blockIdx.x * 256u + threadIdx.x) * 4u;  // pixel index
  size_t b   = p / HW;
  size_t rem = p % HW;
  const float4 r = *(const float4*)(src + (b * 3 + 0) * HW + rem);
  const float4 g = *(const float4*)(src + (b * 3 + 1) * HW + rem);
  const float4 bl = *(const float4*)(src + (b * 3 + 2) * HW + rem);
  float4 o;
  o.x = (r.x + g.x + bl.x) * (1.f / 3.f);
  o.y = (r.y + g.y + bl.y) * (1.f / 3.f);
  o.z = (r.z + g.z + bl.z) * (1.f / 3.f);
  o.w = (r.w + g.w + bl.w) * (1.f / 3.f);
  *(float4*)(dst + b * HW + rem) = o;
}

// ---------------- kernel 2: 2x2 average pool (both images) ----------------
__global__ __launch_bounds__(256) void pool_kernel(
    const float* __restrict__ in0, const float* __restrict__ in1,
    float* __restrict__ out0, float* __restrict__ out1,
    int Hin, int Win) {
  const float* in  = blockIdx.y ? in1 : in0;
  float*       out = blockIdx.y ? out1 : out0;
  const int Hout = Hin >> 1, Wout = Win >> 1;
  const int HWo = Hout * Wout;
  int i = blockIdx.x * 256 + threadIdx.x;        // over 16*Hout*Wout (exact)
  int b = i / HWo;
  int rem = i % HWo;
  int oy = rem / Wout, ox = rem % Wout;
  const float* rp = in + ((size_t)b * Hin + 2 * oy) * Win + 2 * ox;
  float2 a = *(const float2*)rp;
  float2 c = *(const float2*)(rp + Win);
  out[(size_t)b * HWo + rem] = 0.25f * ((a.x + a.y) + (c.x + c.y));
}

// ---------------- kernel 3: per-scale GMS with LDS tiling ----------------
// 32x32 output tile per block (256 threads, 4 outputs/thread).
// Halo region: 36x36 gray values per image (zero-padded at image borders for
// the median, matching the reference's zero-pad median + VALID Prewitt).
__global__ __launch_bounds__(256) void gms_kernel(
    const float* __restrict__ gA, const float* __restrict__ gB,
    int H, int W, int tilesX,
    float* __restrict__ partials, float weight) {
  __shared__ float sGA[36 * 37];
  __shared__ float sGB[36 * 37];
  __shared__ float sMA[34 * 35];
  __shared__ float sMB[34 * 35];
  __shared__ float sRed[8];

  const int tid = threadIdx.x;
  const int b   = blockIdx.y;
  const int ty0 = (blockIdx.x / tilesX) * 32;
  const int tx0 = (blockIdx.x % tilesX) * 32;
  const size_t imgOff = (size_t)b * H * W;
  const float* srcA = gA + imgOff;
  const float* srcB = gB + imgOff;

  // Phase 1: async halo load; zero-fill only out-of-image slots (disjoint
  // addresses -> no DScnt/ASYNCcnt ordering hazard).
  for (int idx = tid; idx < 36 * 36; idx += 256) {
    int r = idx / 36, c = idx % 36;
    int gyy = ty0 - 1 + r;
    int gxx = tx0 - 1 + c;
    int l = r * 37 + c;
    bool ok = ((unsigned)gyy < (unsigned)H) && ((unsigned)gxx < (unsigned)W);
    if (ok) {
      size_t goff = (size_t)gyy * W + gxx;
      async_ld_f32(&sGA[l], srcA + goff);
      async_ld_f32(&sGB[l], srcB + goff);
    } else {
      sGA[l] = 0.f;
      sGB[l] = 0.f;
    }
  }
  wait_async_lds();
  __syncthreads();

  // Phase 2: 34x34 median plane per image.
  for (int idx = tid; idx < 34 * 34; idx += 256) {
    int r = idx / 34, c = idx % 34;
    const float* p = &sGA[r * 37 + c];
    sMA[r * 35 + c] = median9(p[0], p[1], p[2],
                              p[37], p[38], p[39],
                              p[74], p[75], p[76]);
  }
  for (int idx = tid; idx < 34 * 34; idx += 256) {
    int r = idx / 34, c = idx % 34;
    const float* p = &sGB[r * 37 + c];
    sMB[r * 35 + c] = median9(p[0], p[1], p[2],
                              p[37], p[38], p[39],
                              p[74], p[75], p[76]);
  }
  __syncthreads();

  // Phase 3: Prewitt magnitude + GMS term, 4 outputs per thread.
  const int Hout = H - 2, Wout = W - 2;
  float acc = 0.f;
#pragma unroll
  for (int k = 0; k < 4; ++k) {
    int o  = tid + k * 256;
    int oi = o >> 5, oj = o & 31;
    int Oy = ty0 + oi, Ox = tx0 + oj;
    float term = 0.f;
    if (Oy < Hout && Ox < Wout) {
      float gIv = prewitt_mag(&sMA[oi * 35 + oj]);
      float gRv = prewitt_mag(&sMB[oi * 35 + oj]);
      float gmap = (2.f * gIv * gRv + C_GMS) / (gIv * gIv + gRv * gRv + C_GMS);
      term = 1.f - gmap;
    }
    acc += term;
  }

  // Phase 4: deterministic block reduction (WMMA row-sum trick per wave).
  float total = block_sum256(acc, sRed);
  if (tid == 0) {
    partials[(size_t)b * gridDim.x + blockIdx.x] = total * weight;
  }
}

// ---------------- kernel 4: final partial-sum reduction ----------------
__global__ __launch_bounds__(256) void reduce_kernel(
    const float* __restrict__ partials, int n, float* __restrict__ out) {
  __shared__ float sRed[8];
  float acc = 0.f;
  for (int i = threadIdx.x; i < n; i += 256) acc += partials[i];
  float total = block_sum256(acc, sRed);
  if (threadIdx.x == 0) out[0] = total;
}

// ---------------- host launch ----------------
extern "C" void kernel_launch(void* const* d_in, const int* in_sizes, int n_in,
                              void* d_out, int out_size, void* d_ws, size_t ws_size,
                              hipStream_t stream) {
  (void)in_sizes; (void)n_in; (void)out_size;

  const float* Ii = (const float*)d_in[0];   // [16,3,512,512] f32
  const float* Ir = (const float*)d_in[1];   // [16,3,512,512] f32
  float* out = (float*)d_out;                // scalar f32
  float* ws  = (float*)d_ws;                 // needs ~44.6 MB

  const size_t L0 = (size_t)16 * 512 * 512;
  const size_t L1 = (size_t)16 * 256 * 256;
  const size_t L2 = (size_t)16 * 128 * 128;
  const size_t L3 = (size_t)16 * 64 * 64;

  float* gA0 = ws;
  float* gA1 = gA0 + L0;
  float* gA2 = gA1 + L1;
  float* gA3 = gA2 + L2;
  float* gB0 = gA3 + L3;
  float* gB1 = gB0 + L0;
  float* gB2 = gB1 + L1;
  float* gB3 = gB2 + L2;
  float* part = gB3 + L3;                    // 5440 partials
  (void)ws_size;

  // 1) grayscale both images at full resolution
  gray_kernel<<<dim3(4096, 2), 256, 0, stream>>>(Ii, Ir, gA0, gB0);

  // 2) grayscale pyramid (mean over channels commutes with avg-pool)
  pool_kernel<<<dim3(4096, 2), 256, 0, stream>>>(gA0, gB0, gA1, gB1, 512, 512);
  pool_kernel<<<dim3(1024, 2), 256, 0, stream>>>(gA1, gB1, gA2, gB2, 256, 256);
  pool_kernel<<<dim3(256, 2), 256, 0, stream>>>(gA2, gB2, gA3, gB3, 128, 128);

  // 3) per-scale GMS partial sums; weight = 1/(4 * B * (H-2) * (W-2))
  const float w0 = (float)(1.0 / (4.0 * 16.0 * 510.0 * 510.0));
  const float w1 = (float)(1.0 / (4.0 * 16.0 * 254.0 * 254.0));
  const float w2 = (float)(1.0 / (4.0 * 16.0 * 126.0 * 126.0));
  const float w3 = (float)(1.0 / (4.0 * 16.0 * 62.0 * 62.0));

  gms_kernel<<<dim3(256, 16), 256, 0, stream>>>(gA0, gB0, 512, 512, 16, part + 0,    w0);
  gms_kernel<<<dim3(64, 16),  256, 0, stream>>>(gA1, gB1, 256, 256, 8,  part + 4096, w1);
  gms_kernel<<<dim3(16, 16),  256, 0, stream>>>(gA2, gB2, 128, 128, 4,  part + 5120, w2);
  gms_kernel<<<dim3(4, 16),   256, 0, stream>>>(gA3, gB3, 64, 64, 2,    part + 5376, w3);

  // 4) final deterministic reduction to the scalar loss
  reduce_kernel<<<dim3(1), 256, 0, stream>>>(part, 5440, out);
}